// CausalConv1d_89807766159686
// MI455X (gfx1250) — compile-verified
//
#include <hip/hip_runtime.h>
#include <stdint.h>

// Causal depthwise Conv1d, B=4 S=4096 D=2048 K=4, fp32.
// HBM-bandwidth-bound (AI ~= 1 FLOP/byte) -> optimize data movement:
//  - float4 (b128) fully-coalesced loads/stores across the channel dim
//  - CDNA5 async global->LDS staging (ASYNCcnt) for deep prefetch with no
//    VGPR landing-zone pressure
//  - register sliding window along S so every x element is fetched once.

namespace {
constexpr int kB = 4;
constexpr int kS = 4096;
constexpr int kD = 2048;
constexpr int kK = 4;          // taps
constexpr int kTS = 16;        // sequence rows per block tile
constexpr int kThreads = 256;  // 8 wave32
constexpr int kCPB = kThreads * 4;  // 1024 channels per block (float4/thread)
}  // namespace

__global__ __launch_bounds__(kThreads)
void causal_dwconv_f32(const float* __restrict__ x,
                       const float* __restrict__ w,
                       const float* __restrict__ bias,
                       float* __restrict__ out)
{
    // 16 rows x 1024 floats = 64 KiB staging tile (WGP has 320 KiB LDS).
    __shared__ float smem[kTS * kCPB];

    const int tid = threadIdx.x;
    const int d0  = blockIdx.x * kCPB + tid * 4;   // 4 channels per thread
    const int s0  = blockIdx.y * kTS;
    const int b   = blockIdx.z;

    const size_t rowBase = ((size_t)b * kS + s0) * (size_t)kD + (size_t)d0;

    // ---- Stage body rows s0 .. s0+TS-1 into LDS via async b128 loads ----
    // Each lane deposits its own 16 bytes per row; 16 loads in flight per
    // wave, tracked by ASYNCcnt (no VGPR pressure for landing data).
#pragma unroll
    for (int r = 0; r < kTS; ++r) {
        unsigned long long ga =
            (unsigned long long)(uintptr_t)(x + rowBase + (size_t)r * kD);
        unsigned int la =
            (unsigned int)(uintptr_t)(&smem[r * kCPB + tid * 4]);
        asm volatile("global_load_async_to_lds_b128 %0, %1, off"
                     :: "v"(la), "v"(ga)
                     : "memory");
    }

    // ---- Halo rows (s0-3 .. s0-1) directly to registers, zero-guarded ----
    float4 win0, win1, win2;
    {
        const float4 z = make_float4(0.f, 0.f, 0.f, 0.f);
        const float* colBase = x + ((size_t)b * kS) * (size_t)kD + (size_t)d0;
        win0 = (s0 >= 3) ? *(const float4*)(colBase + (size_t)(s0 - 3) * kD) : z;
        win1 = (s0 >= 2) ? *(const float4*)(colBase + (size_t)(s0 - 2) * kD) : z;
        win2 = (s0 >= 1) ? *(const float4*)(colBase + (size_t)(s0 - 1) * kD) : z;
    }

    // ---- Per-channel taps (weight layout [D][1][K], K contiguous) ----
    float4 wc0 = *(const float4*)(w + (size_t)(d0 + 0) * kK);
    float4 wc1 = *(const float4*)(w + (size_t)(d0 + 1) * kK);
    float4 wc2 = *(const float4*)(w + (size_t)(d0 + 2) * kK);
    float4 wc3 = *(const float4*)(w + (size_t)(d0 + 3) * kK);
    // tap-major vectors across the 4 owned channels
    float4 t0 = make_float4(wc0.x, wc1.x, wc2.x, wc3.x);  // multiplies x[s-3]
    float4 t1 = make_float4(wc0.y, wc1.y, wc2.y, wc3.y);  // x[s-2]
    float4 t2 = make_float4(wc0.z, wc1.z, wc2.z, wc3.z);  // x[s-1]
    float4 t3 = make_float4(wc0.w, wc1.w, wc2.w, wc3.w);  // x[s]
    float4 bs = *(const float4*)(bias + d0);

    // Each lane reads only LDS bytes it deposited itself, so its own
    // ASYNCcnt drain is the only ordering needed (no workgroup barrier).
    asm volatile("s_wait_asynccnt 0x0" ::: "memory");

    // ---- Sliding-window compute + coalesced b128 stores ----
#pragma unroll
    for (int r = 0; r < kTS; ++r) {
        float4 xc = *(const float4*)(&smem[r * kCPB + tid * 4]);
        float4 o;
        o.x = fmaf(t3.x, xc.x,
              fmaf(t2.x, win2.x,
              fmaf(t1.x, win1.x,
              fmaf(t0.x, win0.x, bs.x))));
        o.y = fmaf(t3.y, xc.y,
              fmaf(t2.y, win2.y,
              fmaf(t1.y, win1.y,
              fmaf(t0.y, win0.y, bs.y))));
        o.z = fmaf(t3.z, xc.z,
              fmaf(t2.z, win2.z,
              fmaf(t1.z, win1.z,
              fmaf(t0.z, win0.z, bs.z))));
        o.w = fmaf(t3.w, xc.w,
              fmaf(t2.w, win2.w,
              fmaf(t1.w, win1.w,
              fmaf(t0.w, win0.w, bs.w))));
        *(float4*)(out + rowBase + (size_t)r * kD) = o;
        win0 = win1; win1 = win2; win2 = xc;
    }
}

extern "C" void kernel_launch(void* const* d_in, const int* in_sizes, int n_in,
                              void* d_out, int out_size, void* d_ws, size_t ws_size,
                              hipStream_t stream) {
    const float* x    = (const float*)d_in[0];
    const float* w    = (const float*)d_in[1];
    const float* bias = (const float*)d_in[2];
    float* out = (float*)d_out;

    dim3 grid(kD / kCPB, kS / kTS, kB);  // (2, 256, 4)
    causal_dwconv_f32<<<grid, kThreads, 0, stream>>>(x, w, bias, out);

    (void)in_sizes; (void)n_in; (void)out_size; (void)d_ws; (void)ws_size;
}